// DistSAGE_1133871366693
// MI455X (gfx1250) — compile-verified
//
#include <hip/hip_runtime.h>

#define NNODES 100000
#define NEDGES 600000
#define FEATS  128
#define NCLS   64

typedef __attribute__((ext_vector_type(2))) float v2f;
typedef __attribute__((ext_vector_type(8))) float v8f;

// ---------------------------------------------------------------- degree ----
__global__ void deg_count_kernel(const int* __restrict__ dst,
                                 int* __restrict__ deg) {
    int e = blockIdx.x * blockDim.x + threadIdx.x;
    if (e < NEDGES) atomicAdd(&deg[dst[e]], 1);
}

__global__ void inv_deg_kernel(const int* __restrict__ deg,
                               float* __restrict__ invd) {
    int i = blockIdx.x * blockDim.x + threadIdx.x;
    if (i < NNODES) {
        float d = (float)deg[i];
        invd[i] = 1.0f / fmaxf(d, 1.0f);
    }
}

// --------------------------------------------------------------- scatter ----
// One wave (32 lanes) per edge; each lane moves 4 consecutive floats
// (float4 gather from h[src], 4x global_atomic_add_f32 into agg[dst]).
__global__ void scatter_kernel(const float* __restrict__ h,
                               const int* __restrict__ src,
                               const int* __restrict__ dst,
                               float* __restrict__ agg) {
    long long idx = (long long)blockIdx.x * blockDim.x + threadIdx.x;
    int e    = (int)(idx >> 5);
    int lane = (int)(idx & 31);
    if (e >= NEDGES) return;
    int s = src[e];
    int d = dst[e];
    const float4* hp = (const float4*)(h + (long long)s * FEATS);
    float4 v = hp[lane];
    float* ap = agg + (long long)d * FEATS + lane * 4;
    unsafeAtomicAdd(ap + 0, v.x);
    unsafeAtomicAdd(ap + 1, v.y);
    unsafeAtomicAdd(ap + 2, v.z);
    unsafeAtomicAdd(ap + 3, v.w);
}

// ------------------------------------------------------------ WMMA GEMM ----
// out[m,n] = act( hin[m,:] @ Wself[:,n] + (agg[m,:]*invd[m]) @ Wneigh[:,n] + b[n] )
// One workgroup owns a 16-row M-tile; A-tiles (self + invdeg-scaled neigh)
// are staged once into LDS, then each wave computes one 16x16 N-tile with
// V_WMMA_F32_16X16X4_F32 over K=128, reading fragments from LDS.
template <int NCOLS, bool RELU>
__global__ void sage_gemm_kernel(const float* __restrict__ hin,
                                 const float* __restrict__ agg,
                                 const float* __restrict__ invdeg,
                                 const float* __restrict__ Wself,
                                 const float* __restrict__ Wneigh,
                                 const float* __restrict__ bias,
                                 float* __restrict__ out) {
    constexpr int LSTRIDE = FEATS + 4;   // 132 floats: banks = 4*row + ka (conflict-free)
    __shared__ float sSelf[16 * LSTRIDE];
    __shared__ float sNeigh[16 * LSTRIDE];

    const int m0  = blockIdx.x * 16;
    const int tid = threadIdx.y * 32 + threadIdx.x;
    const int nthreads = blockDim.y * 32;

    // cooperative coalesced stage: 16 rows x 128 floats, float4 granularity
    for (int i = tid; i < 16 * (FEATS / 4); i += nthreads) {
        const int row = i >> 5;            // / (FEATS/4)
        const int c4  = i & 31;            // % (FEATS/4)
        const float4* hp = (const float4*)(hin + (long long)(m0 + row) * FEATS);
        const float4* ap = (const float4*)(agg + (long long)(m0 + row) * FEATS);
        float4 hv = hp[c4];
        float4 av = ap[c4];
        const float id = invdeg[m0 + row];
        av.x *= id; av.y *= id; av.z *= id; av.w *= id;
        *(float4*)(sSelf  + row * LSTRIDE + c4 * 4) = hv;
        *(float4*)(sNeigh + row * LSTRIDE + c4 * 4) = av;
    }
    __syncthreads();

    const int lane = threadIdx.x;      // 0..31 (wave32)
    const int half = lane >> 4;        // 0 or 1
    const int l15  = lane & 15;
    const int n0   = threadIdx.y * 16;
    const int n    = n0 + l15;         // B/D column for this lane

    const float* srowS = sSelf  + l15 * LSTRIDE;
    const float* srowN = sNeigh + l15 * LSTRIDE;

    v8f acc = {};   // 16x16 f32 accumulator (8 VGPRs)

    #pragma unroll 4
    for (int k = 0; k < FEATS; k += 4) {
        const int ka = k + half * 2;   // lanes 16..31 hold K=k+2,k+3
        // A fragments (16x4 f32) from LDS: two consecutive K values per lane
        v2f a_s = *(const v2f*)(srowS + ka);
        v2f a_n = *(const v2f*)(srowN + ka);
        // B fragments (4x16 f32): rows ka, ka+1 at column n
        v2f b_s, b_n;
        b_s.x = Wself[(long long)ka * NCOLS + n];
        b_s.y = Wself[(long long)(ka + 1) * NCOLS + n];
        b_n.x = Wneigh[(long long)ka * NCOLS + n];
        b_n.y = Wneigh[(long long)(ka + 1) * NCOLS + n];

        acc = __builtin_amdgcn_wmma_f32_16x16x4_f32(
                  false, a_s, false, b_s, (short)0, acc, false, false);
        acc = __builtin_amdgcn_wmma_f32_16x16x4_f32(
                  false, a_n, false, b_n, (short)0, acc, false, false);
    }

    const float bv = bias[n];
    #pragma unroll
    for (int g = 0; g < 8; ++g) {
        int row = m0 + half * 8 + g;   // D VGPR g: M = g (+8 for upper lanes)
        float v = acc[g] + bv;
        if (RELU) v = fmaxf(v, 0.0f);
        out[(long long)row * NCOLS + n] = v;
    }
}

// ---------------------------------------------------------------- launch ----
static inline size_t align256(size_t x) { return (x + 255) & ~(size_t)255; }

extern "C" void kernel_launch(void* const* d_in, const int* in_sizes, int n_in,
                              void* d_out, int out_size, void* d_ws, size_t ws_size,
                              hipStream_t stream) {
    const float* x   = (const float*)d_in[0];
    const int*   src = (const int*)d_in[1];
    const int*   dst = (const int*)d_in[2];
    const float* Ws0 = (const float*)d_in[3];
    const float* Wn0 = (const float*)d_in[4];
    const float* b0  = (const float*)d_in[5];
    const float* Ws1 = (const float*)d_in[6];
    const float* Wn1 = (const float*)d_in[7];
    const float* b1  = (const float*)d_in[8];
    const float* Ws2 = (const float*)d_in[9];
    const float* Wn2 = (const float*)d_in[10];
    const float* b2  = (const float*)d_in[11];
    float* out = (float*)d_out;

    // workspace carve-out
    char* ws = (char*)d_ws;
    int*   deg  = (int*)ws;          ws += align256((size_t)NNODES * sizeof(int));
    float* invd = (float*)ws;        ws += align256((size_t)NNODES * sizeof(float));
    float* agg  = (float*)ws;        ws += align256((size_t)NNODES * FEATS * sizeof(float));
    float* h1   = (float*)ws;        ws += align256((size_t)NNODES * FEATS * sizeof(float));
    float* h2   = (float*)ws;

    const size_t aggBytes = (size_t)NNODES * FEATS * sizeof(float);
    const int scatterBlocks = (NEDGES * 32) / 256;          // 75000
    const dim3 gemmGrid(NNODES / 16);                       // 6250 M-tiles
    const dim3 gemmBlock128(32, FEATS / 16);                // 8 waves, one N-tile each
    const dim3 gemmBlock64(32, NCLS / 16);                  // 4 waves

    // degrees
    hipMemsetAsync(deg, 0, (size_t)NNODES * sizeof(int), stream);
    deg_count_kernel<<<(NEDGES + 255) / 256, 256, 0, stream>>>(dst, deg);
    inv_deg_kernel<<<(NNODES + 255) / 256, 256, 0, stream>>>(deg, invd);

    // layer 0: x -> h1 (relu)
    hipMemsetAsync(agg, 0, aggBytes, stream);
    scatter_kernel<<<scatterBlocks, 256, 0, stream>>>(x, src, dst, agg);
    sage_gemm_kernel<FEATS, true><<<gemmGrid, gemmBlock128, 0, stream>>>(
        x, agg, invd, Ws0, Wn0, b0, h1);

    // layer 1: h1 -> h2 (relu)
    hipMemsetAsync(agg, 0, aggBytes, stream);
    scatter_kernel<<<scatterBlocks, 256, 0, stream>>>(h1, src, dst, agg);
    sage_gemm_kernel<FEATS, true><<<gemmGrid, gemmBlock128, 0, stream>>>(
        h1, agg, invd, Ws1, Wn1, b1, h2);

    // layer 2: h2 -> out (no activation, N=64)
    hipMemsetAsync(agg, 0, aggBytes, stream);
    scatter_kernel<<<scatterBlocks, 256, 0, stream>>>(h2, src, dst, agg);
    sage_gemm_kernel<NCLS, false><<<gemmGrid, gemmBlock64, 0, stream>>>(
        h2, agg, invd, Ws2, Wn2, b2, out);
}